// CrossNet_2319282340349
// MI455X (gfx1250) — compile-verified
//
#include <hip/hip_runtime.h>

// CrossNet (DCN cross layers), B=16384, D=1024, L=3, fp32.
// One wave32 per row; row kept in VGPRs; w/b staged to LDS via async
// global->LDS loads (CDNA5 ASYNCcnt path); wave-local shfl_xor reductions.

#define DIMS 1024
#define NLAYERS 3
#define CHUNKS 8                 // 8 float4 per lane * 32 lanes * 4 = 1024 floats
#define WAVES_PER_BLOCK 8
#define BLOCK_THREADS (WAVES_PER_BLOCK * 32)
#define VEC_PER_LAYER (DIMS / 4)          // 256 float4 per [D] vector

typedef float f32x4 __attribute__((ext_vector_type(4)));   // native clang vector
typedef int   v4i   __attribute__((vector_size(16)));      // builtin's pointee type
typedef __attribute__((address_space(1))) v4i gv4i;        // global
typedef __attribute__((address_space(3))) v4i lv4i;        // LDS

#if __has_builtin(__builtin_amdgcn_global_load_async_to_lds_b128)
#define USE_ASYNC_LDS 1
#else
#define USE_ASYNC_LDS 0
#endif

__device__ __forceinline__ float dot4(f32x4 a, f32x4 b) {
  f32x4 p = a * b;
  return p.x + p.y + p.z + p.w;
}

__global__ __launch_bounds__(BLOCK_THREADS)
void crossnet_fused(const float* __restrict__ x,
                    const float* __restrict__ w,
                    const float* __restrict__ bias,
                    float* __restrict__ out,
                    int nrows) {
  // LDS layout (f32x4 units): w0[256] w1[256] w2[256] b0[256] b1[256] b2[256]
  __shared__ f32x4 smem4[2 * NLAYERS * VEC_PER_LAYER];     // 24 KB
  const int tid = threadIdx.x;

  // ---- Stage w and b into LDS -------------------------------------------
#if USE_ASYNC_LDS
  {
    const f32x4* gw = (const f32x4*)w;
    const f32x4* gb = (const f32x4*)bias;
#pragma unroll
    for (int i = 0; i < NLAYERS; ++i) {
      const int idx = i * BLOCK_THREADS + tid;             // 0..767
      __builtin_amdgcn_global_load_async_to_lds_b128(
          (gv4i*)(gw + idx), (lv4i*)(smem4 + idx), 0, 0);
      __builtin_amdgcn_global_load_async_to_lds_b128(
          (gv4i*)(gb + idx),
          (lv4i*)(smem4 + NLAYERS * VEC_PER_LAYER + idx), 0, 0);
    }
  }
#else
  {
    const f32x4* gw = (const f32x4*)w;
    const f32x4* gb = (const f32x4*)bias;
    for (int i = tid; i < NLAYERS * VEC_PER_LAYER; i += BLOCK_THREADS) {
      smem4[i] = gw[i];
      smem4[NLAYERS * VEC_PER_LAYER + i] = gb[i];
    }
  }
#endif

  // ---- Load this wave's row of x into registers (overlaps with staging) --
  const int wave = tid >> 5;
  const int lane = tid & 31;
  const long long row = (long long)blockIdx.x * WAVES_PER_BLOCK + wave;
  const bool active = (row < nrows);

  f32x4 x0[CHUNKS];
  if (active) {
    const f32x4* xrow = (const f32x4*)(x + (size_t)row * DIMS);
#pragma unroll
    for (int k = 0; k < CHUNKS; ++k)
      x0[k] = __builtin_nontemporal_load(xrow + k * 32 + lane);
  }

#if USE_ASYNC_LDS
#if __has_builtin(__builtin_amdgcn_s_wait_asynccnt)
  __builtin_amdgcn_s_wait_asynccnt(0);
#else
  asm volatile("s_wait_asynccnt 0" ::: "memory");
#endif
#endif
  __syncthreads();

  if (!active) return;

  const f32x4* wl0 = smem4;
  const f32x4* wl1 = smem4 + 1 * VEC_PER_LAYER;
  const f32x4* wl2 = smem4 + 2 * VEC_PER_LAYER;
  const f32x4* bl0 = smem4 + 3 * VEC_PER_LAYER;
  const f32x4* bl1 = smem4 + 4 * VEC_PER_LAYER;
  const f32x4* bl2 = smem4 + 5 * VEC_PER_LAYER;

  // ---- One pass: three dots d_t = x0.w_t, plus constants c1, c2 ----------
  float d0 = 0.f, d1 = 0.f, d2 = 0.f, c1 = 0.f, c2 = 0.f;
#pragma unroll
  for (int k = 0; k < CHUNKS; ++k) {
    const int idx = k * 32 + lane;
    const f32x4 xv  = x0[k];
    const f32x4 w0v = wl0[idx];
    const f32x4 w1v = wl1[idx];
    const f32x4 w2v = wl2[idx];
    const f32x4 b0v = bl0[idx];
    const f32x4 b1v = bl1[idx];
    d0 += dot4(xv, w0v);
    d1 += dot4(xv, w1v);
    d2 += dot4(xv, w2v);
    c1 += dot4(b0v, w1v);
    c2 += dot4(b0v + b1v, w2v);
  }

  // ---- wave32 butterfly reduction ---------------------------------------
#pragma unroll
  for (int off = 16; off > 0; off >>= 1) {
    d0 += __shfl_xor(d0, off, 32);
    d1 += __shfl_xor(d1, off, 32);
    d2 += __shfl_xor(d2, off, 32);
    c1 += __shfl_xor(c1, off, 32);
    c2 += __shfl_xor(c2, off, 32);
  }

  // ---- Scalar recurrence: xl_t = alpha_t * x0 + sum_{j<t} b_j ------------
  const float s0 = d0;              // c0 == 0
  const float a1 = 1.0f + s0;
  const float s1 = a1 * d1 + c1;
  const float a2 = a1 + s1;
  const float s2 = a2 * d2 + c2;
  const float a3 = a2 + s2;

  // ---- Output pass: out = a3 * x0 + (b0 + b1 + b2) -----------------------
  f32x4* orow = (f32x4*)(out + (size_t)row * DIMS);
#pragma unroll
  for (int k = 0; k < CHUNKS; ++k) {
    const int idx = k * 32 + lane;
    const f32x4 o = a3 * x0[k] + (bl0[idx] + bl1[idx] + bl2[idx]);
    __builtin_nontemporal_store(o, orow + idx);
  }
}

extern "C" void kernel_launch(void* const* d_in, const int* in_sizes, int n_in,
                              void* d_out, int out_size, void* d_ws, size_t ws_size,
                              hipStream_t stream) {
  const float* x = (const float*)d_in[0];     // [B, D]
  const float* w = (const float*)d_in[1];     // [L, D]
  const float* b = (const float*)d_in[2];     // [L, D]
  float* out = (float*)d_out;                 // [B, D]

  const int nrows = in_sizes[0] / DIMS;       // B = 16384
  const int grid = (nrows + WAVES_PER_BLOCK - 1) / WAVES_PER_BLOCK;
  crossnet_fused<<<grid, BLOCK_THREADS, 0, stream>>>(x, w, b, out, nrows);
}